// FastComposerPostfuseModule_10514079940953
// MI455X (gfx1250) — compile-verified
//
#include <hip/hip_runtime.h>
#include <math.h>

// ---------------------------------------------------------------------------
// FastComposer postfuse for MI455X (gfx1250, wave32, WMMA).
//  - copy_kernel : streaming 96MiB copy text->out with non-temporal b128 ops
//  - index_kernel: mask/validity compaction -> pos_for_rank[K], src_obj[K]
//  - fuse_kernel : per-16-row tile, LN -> GEMM(WMMA f32 16x16x4) -> GELU ->
//                  GEMM -> +res -> LN -> GEMM -> GELU -> GEMM -> +res -> LN
//                  -> scatter into out
// ---------------------------------------------------------------------------

#define B_   64
#define S_   512
#define D_   768
#define M_   4
#define T_   1
#define N_   (B_ * S_)       // 32768
#define K_   (B_ * M_ * T_)  // 256
#define TWOD (2 * D_)        // 1536
#define EPSLN 1e-5f

typedef __attribute__((ext_vector_type(2))) float v2f;
typedef __attribute__((ext_vector_type(4))) float vf4;
typedef __attribute__((ext_vector_type(8))) float v8f;

// ---------------------------------------------------------------------------
// Kernel A: streaming copy (dominant cost ~201MB @ 23.3 TB/s). Non-temporal so
// the stream does not evict the L2-resident MLP weights.
// ---------------------------------------------------------------------------
__global__ void copy_kernel(const vf4* __restrict__ src, vf4* __restrict__ dst, int n4) {
    int i = blockIdx.x * blockDim.x + threadIdx.x;
    if (i < n4) {
        vf4 v = __builtin_nontemporal_load(src + i);
        __builtin_nontemporal_store(v, dst + i);
    }
}

// ---------------------------------------------------------------------------
// Kernel B: compute both global compactions (independent cumsums, as in the
// reference). One workgroup; per-batch counts then a tiny serial scan.
// image_token_mask is a bool array -> 1 byte per element.
// ---------------------------------------------------------------------------
__global__ void index_kernel(const unsigned char* __restrict__ mask,
                             const int* __restrict__ num_objects,
                             int* __restrict__ pos_for_rank,
                             int* __restrict__ src_obj) {
    __shared__ int s_mcnt[B_];
    __shared__ int s_mbase[B_];
    __shared__ int s_obase[B_];
    const int tid = threadIdx.x;

    for (int r = tid; r < K_; r += blockDim.x) {
        pos_for_rank[r] = N_;   // sentinel: dropped at scatter
        src_obj[r]      = -1;   // sentinel: zero object row
    }
    if (tid < B_) {
        int c = 0;
        for (int s = 0; s < S_; ++s) c += mask[tid * S_ + s] ? 1 : 0;
        s_mcnt[tid] = c;
    }
    __syncthreads();
    if (tid == 0) {
        int accm = 0, acco = 0;
        for (int b = 0; b < B_; ++b) {
            s_mbase[b] = accm; accm += s_mcnt[b];
            int nv = num_objects[b]; nv = nv < 0 ? 0 : (nv > M_ ? M_ : nv);
            s_obase[b] = acco; acco += nv * T_;
        }
    }
    __syncthreads();
    if (tid < B_) {
        const int b = tid;
        int r = s_mbase[b];
        for (int s = 0; s < S_; ++s) {
            if (mask[b * S_ + s]) { if (r < K_) pos_for_rank[r] = b * S_ + s; ++r; }
        }
        int ro = s_obase[b];
        int nv = num_objects[b]; nv = nv < 0 ? 0 : (nv > M_ ? M_ : nv);
        for (int m = 0; m < nv; ++m)
            for (int t = 0; t < T_; ++t) {
                if (ro < K_) src_obj[ro] = (b * M_ + m) * T_ + t;
                ++ro;
            }
    }
}

// ---------------------------------------------------------------------------
// fp32 WMMA GEMM inner loop. A is a 16-row activation tile in LDS (stride
// LDA), W is the row-major [KDIM, 768] weight in global (L2-resident).
// Per-lane fragment layouts per CDNA5 ISA 7.12.2:
//   A 16x4 : lane L holds rows M=L&15, v2f = K = {koff, koff+1}, koff=2*(L>>4)
//   B 4x16 : lane L holds col N=L&15, v2f = rows {k+koff, k+koff+1}
//   C 16x16: vgpr i, lane L -> M=(L>>4)*8+i, N=n0+(L&15)
// ---------------------------------------------------------------------------
template <int KDIM, int LDA>
__device__ __forceinline__ void wmma_gemm(const float* __restrict__ sA,
                                          const float* __restrict__ W,
                                          int lane, int n0, v8f acc[6]) {
    const int arow = lane & 15;
    const int koff = (lane >> 4) * 2;
    const int col  = lane & 15;
    const float* __restrict__ aBase = sA + arow * LDA + koff;
    #pragma unroll 2
    for (int k = 0; k < KDIM; k += 4) {
        v2f a = *(const v2f*)(aBase + k);                    // 8B-aligned LDS read
        const float* __restrict__ wp = W + (size_t)(k + koff) * D_;
        #pragma unroll
        for (int t = 0; t < 6; ++t) {
            int n = n0 + t * 16 + col;
            v2f b;
            b.x = wp[n];          // row k+koff   (coalesced across lanes 0..15)
            b.y = wp[D_ + n];     // row k+koff+1
            acc[t] = __builtin_amdgcn_wmma_f32_16x16x4_f32(
                false, a, false, b, (short)0, acc[t], false, false);
        }
    }
}

__device__ __forceinline__ float gelu_exact(float x) {
    return 0.5f * x * (1.0f + erff(x * 0.70710678118654752f));
}

// ---------------------------------------------------------------------------
// Kernel C: fused MLP stack for one 16-row tile of the K=256 compact rows.
// 256 threads = 8 wave32; wave w owns output columns [96w, 96w+96).
// LDS plan (~145KB of the 320KB/WGP):
//   s_x  [16][1536] : raw concat -> in-place LN -> reused as
//                     blockA = s_x[0:16*768]      (h1, then y_ln)
//                     blockB = s_x[16*768:16*1536](t2, then y)
//   s_g  [16][768]  : gelu output of GEMM3 (A operand of GEMM4)
// ---------------------------------------------------------------------------
__launch_bounds__(256)
__global__ void fuse_kernel(const float* __restrict__ text,
                            const float* __restrict__ obj,
                            const int* __restrict__ pos_for_rank,
                            const int* __restrict__ src_obj,
                            const float* __restrict__ ln1_w, const float* __restrict__ ln1_b,
                            const float* __restrict__ W1,    const float* __restrict__ b1,
                            const float* __restrict__ W2,    const float* __restrict__ b2,
                            const float* __restrict__ ln2_w, const float* __restrict__ ln2_b,
                            const float* __restrict__ W3,    const float* __restrict__ b3,
                            const float* __restrict__ W4,    const float* __restrict__ b4,
                            const float* __restrict__ lnf_w, const float* __restrict__ lnf_b,
                            float* __restrict__ out) {
    __shared__ float s_x[16 * TWOD];   // 96 KB
    __shared__ float s_g[16 * D_];     // 48 KB
    __shared__ int   s_pos[16];
    __shared__ int   s_src[16];

    float* __restrict__ blockA = s_x;             // 16x768, stride 768
    float* __restrict__ blockB = s_x + 16 * D_;   // 16x768, stride 768

    const int tid  = threadIdx.x;
    const int lane = tid & 31;
    const int wave = tid >> 5;           // 0..7
    const int n0   = wave * 96;
    const int row  = tid >> 4;           // 0..15 (16 threads per row)
    const int t16  = tid & 15;

    if (tid < 16) {
        const int r = blockIdx.x * 16 + tid;
        s_pos[tid] = pos_for_rank[r];
        s_src[tid] = src_obj[r];
    }
    __syncthreads();

    // ---- Phase A: load raw [text | obj] into s_x (clip-gather like take(mode=clip))
    {
        int p = s_pos[row];
        p = p < 0 ? 0 : (p > N_ - 1 ? N_ - 1 : p);
        const vf4* __restrict__ tsrc = (const vf4*)(text + (size_t)p * D_);
        const int sidx = s_src[row];
        const vf4* __restrict__ osrc =
            (sidx >= 0) ? (const vf4*)(obj + (size_t)sidx * D_) : nullptr;
        for (int j = t16; j < D_ / 4; j += 16) {
            vf4 tv = tsrc[j];
            *(vf4*)&s_x[row * TWOD + 4 * j] = tv;
            vf4 ov = {0.f, 0.f, 0.f, 0.f};
            if (osrc) ov = osrc[j];
            *(vf4*)&s_x[row * TWOD + D_ + 4 * j] = ov;
        }
    }
    __syncthreads();

    // ---- Phase B: LayerNorm over 1536 (in place), 16-lane shuffle reduction
    {
        float sum = 0.f, sq = 0.f;
        for (int j = t16; j < TWOD; j += 16) {
            float v = s_x[row * TWOD + j];
            sum += v; sq += v * v;
        }
        for (int o = 8; o >= 1; o >>= 1) {
            sum += __shfl_xor(sum, o, 16);
            sq  += __shfl_xor(sq,  o, 16);
        }
        const float mu = sum * (1.f / TWOD);
        const float rs = rsqrtf(sq * (1.f / TWOD) - mu * mu + EPSLN);
        for (int j = t16; j < TWOD; j += 16) {
            float v = s_x[row * TWOD + j];
            s_x[row * TWOD + j] = (v - mu) * rs * ln1_w[j] + ln1_b[j];
        }
    }
    __syncthreads();

    const int mbase = (lane >> 4) * 8;
    const int ncol  = lane & 15;
    v8f acc[6];

    // ---- Phase C: GEMM1 [16,1536] x [1536,768], then GELU -> blockA
    #pragma unroll
    for (int t = 0; t < 6; ++t) { v8f z = {0,0,0,0,0,0,0,0}; acc[t] = z; }
    wmma_gemm<TWOD, TWOD>(s_x, W1, lane, n0, acc);
    __syncthreads();   // all GEMM1 reads of s_x done before overwriting blockA
    #pragma unroll
    for (int t = 0; t < 6; ++t) {
        const int n = n0 + t * 16 + ncol;
        const float bias = b1[n];
        #pragma unroll
        for (int i = 0; i < 8; ++i) {
            const int m = mbase + i;
            blockA[m * D_ + n] = gelu_exact(acc[t][i] + bias);
        }
    }
    __syncthreads();

    // ---- Phase D: GEMM2 [16,768] x [768,768], + text residual -> t2 in blockB
    #pragma unroll
    for (int t = 0; t < 6; ++t) { v8f z = {0,0,0,0,0,0,0,0}; acc[t] = z; }
    wmma_gemm<D_, D_>(blockA, W2, lane, n0, acc);
    #pragma unroll
    for (int t = 0; t < 6; ++t) {
        const int n = n0 + t * 16 + ncol;
        const float bias = b2[n];
        #pragma unroll
        for (int i = 0; i < 8; ++i) {
            const int m = mbase + i;
            int p = s_pos[m];
            p = p < 0 ? 0 : (p > N_ - 1 ? N_ - 1 : p);
            const float tv = text[(size_t)p * D_ + n];   // residual (L2-hot)
            blockB[m * D_ + n] = acc[t][i] + bias + tv;
        }
    }
    __syncthreads();

    // ---- Phase E: LayerNorm(t2) over 768 -> y_ln in blockA
    {
        float sum = 0.f, sq = 0.f;
        for (int j = t16; j < D_; j += 16) {
            float v = blockB[row * D_ + j];
            sum += v; sq += v * v;
        }
        for (int o = 8; o >= 1; o >>= 1) {
            sum += __shfl_xor(sum, o, 16);
            sq  += __shfl_xor(sq,  o, 16);
        }
        const float mu = sum * (1.f / D_);
        const float rs = rsqrtf(sq * (1.f / D_) - mu * mu + EPSLN);
        for (int j = t16; j < D_; j += 16) {
            float v = blockB[row * D_ + j];
            blockA[row * D_ + j] = (v - mu) * rs * ln2_w[j] + ln2_b[j];
        }
    }
    __syncthreads();

    // ---- Phase F: GEMM3, GELU -> s_g
    #pragma unroll
    for (int t = 0; t < 6; ++t) { v8f z = {0,0,0,0,0,0,0,0}; acc[t] = z; }
    wmma_gemm<D_, D_>(blockA, W3, lane, n0, acc);
    #pragma unroll
    for (int t = 0; t < 6; ++t) {
        const int n = n0 + t * 16 + ncol;
        const float bias = b3[n];
        #pragma unroll
        for (int i = 0; i < 8; ++i) {
            const int m = mbase + i;
            s_g[m * D_ + n] = gelu_exact(acc[t][i] + bias);
        }
    }
    __syncthreads();

    // ---- Phase G: GEMM4, + t2 residual -> y in blockB (owner overwrite)
    #pragma unroll
    for (int t = 0; t < 6; ++t) { v8f z = {0,0,0,0,0,0,0,0}; acc[t] = z; }
    wmma_gemm<D_, D_>(s_g, W4, lane, n0, acc);
    #pragma unroll
    for (int t = 0; t < 6; ++t) {
        const int n = n0 + t * 16 + ncol;
        const float bias = b4[n];
        #pragma unroll
        for (int i = 0; i < 8; ++i) {
            const int m = mbase + i;
            blockB[m * D_ + n] = acc[t][i] + bias + blockB[m * D_ + n];
        }
    }
    __syncthreads();

    // ---- Phase H: final LayerNorm + scatter into out (drop sentinel rows)
    {
        float sum = 0.f, sq = 0.f;
        for (int j = t16; j < D_; j += 16) {
            float v = blockB[row * D_ + j];
            sum += v; sq += v * v;
        }
        for (int o = 8; o >= 1; o >>= 1) {
            sum += __shfl_xor(sum, o, 16);
            sq  += __shfl_xor(sq,  o, 16);
        }
        const float mu = sum * (1.f / D_);
        const float rs = rsqrtf(sq * (1.f / D_) - mu * mu + EPSLN);
        const int p = s_pos[row];
        if (p >= 0 && p < N_) {
            for (int j = t16; j < D_; j += 16) {
                float v = blockB[row * D_ + j];
                out[(size_t)p * D_ + j] = (v - mu) * rs * lnf_w[j] + lnf_b[j];
            }
        }
    }
}

// ---------------------------------------------------------------------------
extern "C" void kernel_launch(void* const* d_in, const int* in_sizes, int n_in,
                              void* d_out, int out_size, void* d_ws, size_t ws_size,
                              hipStream_t stream) {
    const float* text          = (const float*)d_in[0];
    const float* obj           = (const float*)d_in[1];
    const unsigned char* mask  = (const unsigned char*)d_in[2];  // bool array
    const int*   num_objects   = (const int*)d_in[3];
    const float* ln1_w = (const float*)d_in[4];
    const float* ln1_b = (const float*)d_in[5];
    const float* W1    = (const float*)d_in[6];
    const float* b1    = (const float*)d_in[7];
    const float* W2    = (const float*)d_in[8];
    const float* b2    = (const float*)d_in[9];
    const float* ln2_w = (const float*)d_in[10];
    const float* ln2_b = (const float*)d_in[11];
    const float* W3    = (const float*)d_in[12];
    const float* b3    = (const float*)d_in[13];
    const float* W4    = (const float*)d_in[14];
    const float* b4    = (const float*)d_in[15];
    const float* lnf_w = (const float*)d_in[16];
    const float* lnf_b = (const float*)d_in[17];
    float* out = (float*)d_out;

    int* pos_for_rank = (int*)d_ws;
    int* src_obj      = pos_for_rank + K_;

    index_kernel<<<1, 256, 0, stream>>>(mask, num_objects, pos_for_rank, src_obj);

    const int n4 = N_ * D_ / 4;   // 6,291,456 float4
    copy_kernel<<<(n4 + 255) / 256, 256, 0, stream>>>((const vf4*)text, (vf4*)out, n4);

    fuse_kernel<<<K_ / 16, 256, 0, stream>>>(
        text, obj, pos_for_rank, src_obj,
        ln1_w, ln1_b, W1, b1, W2, b2,
        ln2_w, ln2_b, W3, b3, W4, b4,
        lnf_w, lnf_b, out);
}